// GATLayer_14809047237107
// MI455X (gfx1250) — compile-verified
//
#include <hip/hip_runtime.h>
#include <hip/hip_bf16.h>

typedef float v2f __attribute__((ext_vector_type(2)));
typedef float v8f __attribute__((ext_vector_type(8)));

#define H_HEADS 4
#define HF 128
#define F_IN 128
#define EDGE_DIM 64
#define HIDDEN 256
#define LN_EPS 1e-5f

static __device__ __forceinline__ v8f wmma_f32(v2f a, v2f b, v8f c) {
  // V_WMMA_F32_16X16X4_F32: (neg_a, A, neg_b, B, c_mod, C, reuse_a, reuse_b)
  return __builtin_amdgcn_wmma_f32_16x16x4_f32(false, a, false, b, (short)0, c,
                                               false, false);
}

static __device__ __forceinline__ unsigned f2ord(float f) {
  unsigned u = __float_as_uint(f);
  return (u & 0x80000000u) ? ~u : (u | 0x80000000u);
}
static __device__ __forceinline__ float ord2f(unsigned u) {
  unsigned b = (u & 0x80000000u) ? (u ^ 0x80000000u) : ~u;
  return __uint_as_float(b);
}

// ---------------- instruction bridges: [B,256] @ [256,128] + bias, x3 -------
__global__ void __launch_bounds__(128)
k_bridges(const float* __restrict__ ins,
          const float* __restrict__ Wsrc, const float* __restrict__ bsrc,
          const float* __restrict__ Wtrg, const float* __restrict__ btrg,
          const float* __restrict__ Wedg, const float* __restrict__ bedg,
          float* __restrict__ br_src, float* __restrict__ br_trg,
          float* __restrict__ br_edg) {
  int b = blockIdx.x;
  int c = threadIdx.x;
  const float* irow = ins + (size_t)b * HIDDEN;
  float s0 = 0.f, s1 = 0.f, s2 = 0.f;
  for (int k = 0; k < HIDDEN; ++k) {
    float iv = irow[k];
    s0 += iv * Wsrc[(size_t)k * HF + c];
    s1 += iv * Wtrg[(size_t)k * HF + c];
    s2 += iv * Wedg[(size_t)k * HF + c];
  }
  br_src[(size_t)b * HF + c] = s0 + bsrc[c];
  br_trg[(size_t)b * HF + c] = s1 + btrg[c];
  br_edg[(size_t)b * HF + c] = s2 + bedg[c];
}

// ---------------- proj & skip GEMMs via fp32 WMMA ---------------------------
__global__ void __launch_bounds__(128)
k_proj_skip(const float* __restrict__ x, const float* __restrict__ Wp,
            const float* __restrict__ Ws, float* __restrict__ proj,
            float* __restrict__ skip, int n_row_tiles) {
  int wave = threadIdx.x >> 5;
  int lane = threadIdx.x & 31;
  int tile = blockIdx.x * 4 + wave;
  int rt = tile >> 3;  // row tile (16 nodes)
  int ct = tile & 7;   // col tile (16 of 128 outputs)
  if (rt >= n_row_tiles) return;  // wave-uniform
  int half = lane >> 4;
  int lm = lane & 15;
  int row = rt * 16 + lm;
  int col = ct * 16 + lm;
  v8f accp = {};
  v8f accs = {};
  const float* xrow = x + (size_t)row * F_IN;
  for (int kb = 0; kb < F_IN; kb += 4) {
    int k = kb + 2 * half;
    v2f a = *(const v2f*)(xrow + k);  // A frag: row M, K..K+1
    v2f bp, bs;
    bp.x = Wp[(size_t)k * HF + col];
    bp.y = Wp[(size_t)(k + 1) * HF + col];
    bs.x = Ws[(size_t)k * HF + col];
    bs.y = Ws[(size_t)(k + 1) * HF + col];
    accp = wmma_f32(a, bp, accp);
    accs = wmma_f32(a, bs, accs);
  }
  for (int i = 0; i < 8; ++i) {
    int r = rt * 16 + i + 8 * half;
    proj[(size_t)r * HF + col] = accp[i];
    skip[(size_t)r * HF + col] = accs[i];
  }
}

// ------------- eproj GEMM (WMMA) with fused per-edge score reduction --------
__global__ void __launch_bounds__(128)
k_eproj_scores(const float* __restrict__ edges, const float* __restrict__ We,
               const float* __restrict__ a_edge, const float* __restrict__ ebr,
               const int* __restrict__ batch_ids, float* __restrict__ se,
               int n_row_tiles) {
  int wave = threadIdx.x >> 5;
  int lane = threadIdx.x & 31;
  int tile = blockIdx.x * 4 + wave;
  int rt = tile >> 3;
  int ct = tile & 7;
  if (rt >= n_row_tiles) return;  // wave-uniform
  int half = lane >> 4;
  int lm = lane & 15;
  int row = rt * 16 + lm;
  int col = ct * 16 + lm;
  v8f acc = {};
  const float* erow = edges + (size_t)row * EDGE_DIM;
  for (int kb = 0; kb < EDGE_DIM; kb += 4) {
    int k = kb + 2 * half;
    v2f a = *(const v2f*)(erow + k);
    v2f b;
    b.x = We[(size_t)k * HF + col];
    b.y = We[(size_t)(k + 1) * HF + col];
    acc = wmma_f32(a, b, acc);
  }
  int h = ct >> 1;  // 32 cols per head, 16 per tile
  float ae = a_edge[col];
  for (int i = 0; i < 8; ++i) {
    int r = rt * 16 + i + 8 * half;
    int b = batch_ids[r];
    float v = acc[i] * ae * ebr[(size_t)b * HF + col];
    // reduce over 16 lanes of this half-wave
    for (int m = 8; m >= 1; m >>= 1) v += __shfl_xor(v, m, 16);
    if (lm == 0) atomicAdd(&se[(size_t)r * H_HEADS + h], v);
  }
}

// ---------------- per-node attention scores ---------------------------------
__global__ void __launch_bounds__(128)
k_node_scores(const float* __restrict__ proj, const float* __restrict__ a_src,
              const float* __restrict__ a_trg, const float* __restrict__ br_src,
              const float* __restrict__ br_trg, const int* __restrict__ mle_ptr,
              float* __restrict__ ss, float* __restrict__ st) {
  int n = blockIdx.x;
  int c = threadIdx.x;  // 0..127 ; each wave == one head
  int mle = *mle_ptr;
  int b = n / mle;
  float p = proj[(size_t)n * HF + c];
  float vs = p * a_src[c] * br_src[(size_t)b * HF + c];
  float vt = p * a_trg[c] * br_trg[(size_t)b * HF + c];
  for (int m = 16; m >= 1; m >>= 1) {
    vs += __shfl_xor(vs, m, 32);
    vt += __shfl_xor(vt, m, 32);
  }
  if ((c & 31) == 0) {
    int h = c >> 5;
    ss[(size_t)n * H_HEADS + h] = vs;
    st[(size_t)n * H_HEADS + h] = vt;
  }
}

// ---------------- edge logits (leaky-relu) + global max ---------------------
__global__ void __launch_bounds__(256)
k_edge_logits(const int* __restrict__ eidx, const float* __restrict__ ss,
              const float* __restrict__ st, const float* __restrict__ se,
              float* __restrict__ spe, unsigned* __restrict__ gmax,
              long long total, long long E) {
  long long idx = (long long)blockIdx.x * 256 + threadIdx.x;
  float s = -3.4e38f;
  if (idx < total) {
    long long e = idx >> 2;
    int h = (int)(idx & 3);
    int srcn = eidx[e];
    int trgn = eidx[E + e];
    float x = ss[(size_t)srcn * H_HEADS + h] + st[(size_t)trgn * H_HEADS + h] +
              se[idx];
    s = x > 0.f ? x : 0.2f * x;
    spe[idx] = s;
  }
  for (int m = 16; m >= 1; m >>= 1) s = fmaxf(s, __shfl_xor(s, m, 32));
  __shared__ float sm[8];
  int wv = threadIdx.x >> 5;
  if ((threadIdx.x & 31) == 0) sm[wv] = s;
  __syncthreads();
  if (threadIdx.x == 0) {
    float m = sm[0];
    for (int i = 1; i < 8; ++i) m = fmaxf(m, sm[i]);
    atomicMax(gmax, f2ord(m));
  }
}

// ---------------- exp + segment-sum denominators (both directions) ----------
__global__ void __launch_bounds__(256)
k_denom(const int* __restrict__ eidx, float* __restrict__ spe,
        const unsigned* __restrict__ gmax, float* __restrict__ den_src,
        float* __restrict__ den_trg, long long total, long long E) {
  long long idx = (long long)blockIdx.x * 256 + threadIdx.x;
  if (idx >= total) return;
  float gm = ord2f(*gmax);
  long long e = idx >> 2;
  int h = (int)(idx & 3);
  float ex = __expf(spe[idx] - gm);
  spe[idx] = ex;  // keep exp in place for scatter pass
  int srcn = eidx[e];
  int trgn = eidx[E + e];
  atomicAdd(&den_trg[(size_t)trgn * H_HEADS + h], ex);
  atomicAdd(&den_src[(size_t)srcn * H_HEADS + h], ex);
}

// ---------------- weighted scatter into output accumulators -----------------
__global__ void __launch_bounds__(128)
k_scatter(const int* __restrict__ eidx, const float* __restrict__ proj,
          const float* __restrict__ spe, const float* __restrict__ den_src,
          const float* __restrict__ den_trg, float* __restrict__ out,
          long long E) {
  long long e = blockIdx.x;
  int c = threadIdx.x;  // 0..127
  int h = c >> 5;
  int srcn = eidx[e];
  int trgn = eidx[E + e];
  float ex = spe[e * 4 + h];
  float att_t = ex / (den_trg[(size_t)trgn * H_HEADS + h] + 1e-16f);
  float att_s = ex / (den_src[(size_t)srcn * H_HEADS + h] + 1e-16f);
  // trg_feats: lift src -> agg trg (cols 128..255)
  atomicAdd(&out[(size_t)trgn * 256 + 128 + c],
            proj[(size_t)srcn * HF + c] * att_t);
  // src_feats: lift trg -> agg src (cols 0..127)
  atomicAdd(&out[(size_t)srcn * 256 + c],
            proj[(size_t)trgn * HF + c] * att_s);
}

// ---------------- skip + bias + LayerNorm(256) in place ---------------------
__global__ void __launch_bounds__(256)
k_finalize(float* __restrict__ out, const float* __restrict__ skip,
           const float* __restrict__ bias, const float* __restrict__ gamma,
           const float* __restrict__ beta) {
  int n = blockIdx.x;
  int c = threadIdx.x;  // 0..255
  float v = out[(size_t)n * 256 + c] + skip[(size_t)n * HF + (c & 127)] +
            bias[c & 127];
  float s = v, s2 = v * v;
  for (int m = 16; m >= 1; m >>= 1) {
    s += __shfl_xor(s, m, 32);
    s2 += __shfl_xor(s2, m, 32);
  }
  __shared__ float sm[8], sm2[8];
  int wv = threadIdx.x >> 5;
  if ((threadIdx.x & 31) == 0) {
    sm[wv] = s;
    sm2[wv] = s2;
  }
  __syncthreads();
  float ts = 0.f, ts2 = 0.f;
  for (int i = 0; i < 8; ++i) {
    ts += sm[i];
    ts2 += sm2[i];
  }
  float mu = ts * (1.f / 256.f);
  float var = ts2 * (1.f / 256.f) - mu * mu;
  out[(size_t)n * 256 + c] = (v - mu) * rsqrtf(var + LN_EPS) * gamma[c] + beta[c];
}

extern "C" void kernel_launch(void* const* d_in, const int* in_sizes, int n_in,
                              void* d_out, int out_size, void* d_ws,
                              size_t ws_size, hipStream_t stream) {
  (void)n_in; (void)out_size; (void)ws_size;
  const float* x      = (const float*)d_in[0];
  const int*   eidx   = (const int*)d_in[1];
  const float* edges  = (const float*)d_in[2];
  const float* ins    = (const float*)d_in[3];
  const int*   bids   = (const int*)d_in[4];
  const int*   mle    = (const int*)d_in[5];
  const float* Wp     = (const float*)d_in[6];
  const float* We     = (const float*)d_in[7];
  const float* Wsi    = (const float*)d_in[8];
  const float* bsi    = (const float*)d_in[9];
  const float* Wti    = (const float*)d_in[10];
  const float* bti    = (const float*)d_in[11];
  const float* Wei    = (const float*)d_in[12];
  const float* bei    = (const float*)d_in[13];
  const float* a_src  = (const float*)d_in[14];
  const float* a_trg  = (const float*)d_in[15];
  const float* a_edge = (const float*)d_in[16];
  const float* bias   = (const float*)d_in[17];
  const float* Wskip  = (const float*)d_in[18];
  const float* gamma  = (const float*)d_in[19];
  const float* beta   = (const float*)d_in[20];

  long long N = in_sizes[0] / F_IN;       // 96000
  long long E = in_sizes[2] / EDGE_DIM;   // 600000
  long long B = in_sizes[3] / HIDDEN;     // 64
  float* out = (float*)d_out;

  char* ws = (char*)d_ws;
  size_t off = 0;
  auto alloc = [&](size_t nfloats) {
    float* p = (float*)(ws + off);
    off += nfloats * sizeof(float);
    return p;
  };
  float* proj = alloc((size_t)N * HF);
  float* skip = alloc((size_t)N * HF);
  float* ss   = alloc((size_t)N * H_HEADS);
  float* st   = alloc((size_t)N * H_HEADS);
  float* se   = alloc((size_t)E * H_HEADS);
  float* spe  = alloc((size_t)E * H_HEADS);
  float* dsrc = alloc((size_t)N * H_HEADS);
  float* dtrg = alloc((size_t)N * H_HEADS);
  float* brs  = alloc((size_t)B * HF);
  float* brt  = alloc((size_t)B * HF);
  float* bre  = alloc((size_t)B * HF);
  unsigned* gmax = (unsigned*)(ws + off);

  hipMemsetAsync(out, 0, (size_t)N * 256 * sizeof(float), stream);
  hipMemsetAsync(se, 0, (size_t)E * H_HEADS * sizeof(float), stream);
  hipMemsetAsync(dsrc, 0, (size_t)N * H_HEADS * sizeof(float), stream);
  hipMemsetAsync(dtrg, 0, (size_t)N * H_HEADS * sizeof(float), stream);
  hipMemsetAsync(gmax, 0, sizeof(unsigned), stream);

  k_bridges<<<(int)B, 128, 0, stream>>>(ins, Wsi, bsi, Wti, bti, Wei, bei,
                                        brs, brt, bre);
  int nrt = (int)(N / 16);
  k_proj_skip<<<nrt * 2, 128, 0, stream>>>(x, Wp, Wskip, proj, skip, nrt);
  int ert = (int)(E / 16);
  k_eproj_scores<<<ert * 2, 128, 0, stream>>>(edges, We, a_edge, bre, bids,
                                              se, ert);
  k_node_scores<<<(int)N, 128, 0, stream>>>(proj, a_src, a_trg, brs, brt, mle,
                                            ss, st);
  long long total = E * H_HEADS;
  int blks = (int)((total + 255) / 256);
  k_edge_logits<<<blks, 256, 0, stream>>>(eidx, ss, st, se, spe, gmax, total, E);
  k_denom<<<blks, 256, 0, stream>>>(eidx, spe, gmax, dsrc, dtrg, total, E);
  k_scatter<<<(int)E, 128, 0, stream>>>(eidx, proj, spe, dsrc, dtrg, out, E);
  k_finalize<<<(int)N, 256, 0, stream>>>(out, skip, bias, gamma, beta);
}